// s4_kernel_72395968741804
// MI455X (gfx1250) — compile-verified
//
#include <hip/hip_runtime.h>

#define SSM_N 256
#define SEQ_L 65536
#define A_PAD 2   // row stride SSM_N+2 float2s -> 516 words, == 4 mod 64: bank-conflict-free

typedef float v2f __attribute__((ext_vector_type(2)));
typedef float v8f __attribute__((ext_vector_type(8)));

// Fast reciprocal: v_rcp_f32 + one Newton-Raphson step (~1 ulp) instead of the
// IEEE v_div_scale/v_div_fmas/v_div_fixup sequence (~10 dependent VALU ops).
__device__ __forceinline__ float fast_rcp(float d) {
    float r = __builtin_amdgcn_rcpf(d);
    float e = __builtin_fmaf(-d, r, 1.0f);
    return __builtin_fmaf(r, e, r);
}

// Each wave computes 16 consecutive frequency nodes l.
// Real-valued formulation of the complex [4,N]@[N,16] Cauchy matmul:
//   A (16 x 2N): rows 0-3 -> Re(K_k), rows 4-7 -> Im(K_k), rows 8-15 zero pad
//       A[k,   2n] =  vs_r[k,n],  A[k,   2n+1] = -vs_i[k,n]
//       A[4+k, 2n] =  vs_i[k,n],  A[4+k, 2n+1] =  vs_r[k,n]
//   B (2N x 16): B[2n,l] = Re(1/(g_l - Lam_n)),  B[2n+1,l] = Im(...)
// Accumulated with 128x V_WMMA_F32_16X16X4_F32 per wave (K-chunk = 2 poles).
// A is pre-materialized per (row,n) in LDS as float2 so the hot loop is:
// 2x ds_load_b64 + reciprocal math + v_wmma (VALU co-executes with XDL).
__global__ __launch_bounds__(128) void s4_cauchy_wmma_kernel(
    const float* __restrict__ Lam,     // [N] complex64 interleaved
    const float* __restrict__ Pv,
    const float* __restrict__ Qv,
    const float* __restrict__ Bv,
    const float* __restrict__ Cv,
    const float* __restrict__ stepPtr, // scalar
    float* __restrict__ out)           // [L] complex64 interleaved
{
    __shared__ float2 s_a[16][SSM_N + A_PAD];  // A-operand pairs per (row, n)
    __shared__ float2 s_lam[SSM_N];            // Lambda as (re, im)

    const int tid = threadIdx.x;

    // Stage Lambda.
    for (int n = tid; n < SSM_N; n += 128) {
        float2 lm;
        lm.x = Lam[2*n];
        lm.y = Lam[2*n+1];
        s_lam[n] = lm;
    }

    // Stage realified A: 16 rows x 256 poles. vs_k = conj(x_k)*y_k with
    // x in {C,C,Q,Q}, y in {B,P,B,P}.
    for (int idx = tid; idx < 16 * SSM_N; idx += 128) {
        const int row = idx & 15;
        const int n   = idx >> 4;
        const int k   = row & 3;
        const float* xs = (k < 2)  ? Cv : Qv;
        const float* ys = (k & 1) ? Pv : Bv;
        const float xr = xs[2*n], xi = xs[2*n+1];
        const float yr = ys[2*n], yi = ys[2*n+1];
        const float vr = xr*yr + xi*yi;   // Re(conj(x)*y)
        const float vi = xr*yi - xi*yr;   // Im(conj(x)*y)
        float2 a;
        if (row < 4)      { a.x = vr;   a.y = -vi; }
        else if (row < 8) { a.x = vi;   a.y =  vr; }
        else              { a.x = 0.f;  a.y = 0.f; }
        s_a[row][n] = a;
    }
    __syncthreads();

    const float two_over_step = 2.0f / stepPtr[0];

    const int wave  = tid >> 5;
    const int lane  = tid & 31;
    const int col   = lane & 15;   // frequency within the wave's 16-wide tile
    const int lhalf = lane >> 4;   // which pole of the K-chunk this lane feeds

    const int l = (blockIdx.x * 4 + wave) * 16 + col;

    // Omega = exp(-2*pi*i*l/L) in fp32.
    const float theta = -6.28318530717958647692f * ((float)l / (float)SEQ_L);
    const float omr = __cosf(theta);
    const float omi = __sinf(theta);

    // c = 2/(1+Omega)
    const float dr = 1.0f + omr, di = omi;
    const float invd = fast_rcp(dr*dr + di*di);
    const float cfr =  2.0f * dr * invd;
    const float cfi = -2.0f * di * invd;

    // g = (2/step)*(1-Omega)/(1+Omega), full complex (tiny real part kept).
    const float nr = 1.0f - omr, ni = -omi;
    const float gr = two_over_step * (nr*dr + ni*di) * invd;
    const float gi = two_over_step * (ni*dr - nr*di) * invd;

    // Per-lane loop-invariant A-row pointer (row = col, 0..15).
    const float2* __restrict__ pa = &s_a[col][0];

    v8f acc = {0.f, 0.f, 0.f, 0.f, 0.f, 0.f, 0.f, 0.f};

#pragma unroll 4
    for (int n0 = 0; n0 < SSM_N; n0 += 2) {
        const int n = n0 + lhalf;

        // B operand: complex reciprocal 1/(g - Lambda[n]) for column `col`.
        const float2 lm = s_lam[n];
        const float ar  = gr - lm.x;
        const float ai  = gi - lm.y;
        const float idn = fast_rcp(__builtin_fmaf(ar, ar, ai*ai));
        v2f b;
        b.x =  ar * idn;
        b.y = -ai * idn;

        // A operand straight from LDS (bank-conflict-free b64 load).
        const float2 av = pa[n];
        v2f a;
        a.x = av.x;
        a.y = av.y;

        // D = A x B + C ; fp32 WMMA accumulate.
        acc = __builtin_amdgcn_wmma_f32_16x16x4_f32(
            /*neg_a=*/false, a, /*neg_b=*/false, b,
            /*c_mod=*/(short)0, acc, /*reuse_a=*/false, /*reuse_b=*/false);
    }

    // Lanes 0-15 hold rows 0-7 of D: k00r..k11r in acc[0..3], k00i..k11i in acc[4..7].
    if (lhalf == 0) {
        const float k00r = acc[0], k01r = acc[1], k10r = acc[2], k11r = acc[3];
        const float k00i = acc[4], k01i = acc[5], k10i = acc[6], k11i = acc[7];

        // atRoots = c * (k00 - k01 * k10 / (1 + k11))
        const float tr  = 1.0f + k11r, ti = k11i;
        const float itn = fast_rcp(tr*tr + ti*ti);
        const float ur  = k01r*k10r - k01i*k10i;
        const float ui  = k01r*k10i + k01i*k10r;
        const float wr  = (ur*tr + ui*ti) * itn;
        const float wi  = (ui*tr - ur*ti) * itn;
        const float zr  = k00r - wr;
        const float zi  = k00i - wi;

        float2 res;
        res.x = cfr*zr - cfi*zi;
        res.y = cfr*zi + cfi*zr;
        ((float2*)out)[l] = res;
    }
}

extern "C" void kernel_launch(void* const* d_in, const int* in_sizes, int n_in,
                              void* d_out, int out_size, void* d_ws, size_t ws_size,
                              hipStream_t stream) {
    const float* Lam = (const float*)d_in[0];
    const float* Pv  = (const float*)d_in[1];
    const float* Qv  = (const float*)d_in[2];
    const float* Bv  = (const float*)d_in[3];
    const float* Cv  = (const float*)d_in[4];
    const float* st  = (const float*)d_in[5];
    float* out = (float*)d_out;

    // 4 waves/block * 16 frequencies/wave = 64 frequencies per block.
    dim3 grid(SEQ_L / 64);
    dim3 block(128);
    s4_cauchy_wmma_kernel<<<grid, block, 0, stream>>>(Lam, Pv, Qv, Bv, Cv, st, out);
}